// cooperative_pretraining_baseline_shape_context_prediction_loss_12043088298401
// MI455X (gfx1250) — compile-verified
//
#include <hip/hip_runtime.h>
#include <math.h>

// Problem constants (match reference)
#define NQ     1024      // vehicle points
#define NF     20000     // fusion points
#define TOPK_  200
#define NTILES (NF / 16) // 1250, exact
#define NWAVES 8         // 256 threads / wave32

typedef __attribute__((ext_vector_type(2))) float v2f;
typedef __attribute__((ext_vector_type(8))) float v8f;

// One V_WMMA_F32_16X16X4_F32 produces a 16x16 tile of squared distances:
//   A_m  = (x, y, z, 1)                (rows = 16 vehicle points of this WG)
//   B_n  = (-2fx, -2fy, -2fz, |f|^2)^T (cols = 16 fusion points of the tile)
//   C_mn = |a_m|^2
//   D = A*B + C = |a|^2 - 2 a.f + |f|^2
__device__ __forceinline__ v8f d2_tile(const float* __restrict__ fus, int tileBase,
                                       v2f afrag, v8f cfrag, bool hi, int n) {
    const float* fp = fus + 3 * (tileBase + n);
    const float fx = fp[0], fy = fp[1], fz = fp[2];
    const float n2 = fx * fx + fy * fy + fz * fz;  // unconditional -> cndmask, not exec branches
    v2f b;
    // B (KxN, 4x16): lanes 0-15 hold K={0,1}, lanes 16-31 hold K={2,3}
    b.x = hi ? (-2.0f * fz) : (-2.0f * fx);
    b.y = hi ? n2 : (-2.0f * fy);
    return __builtin_amdgcn_wmma_f32_16x16x4_f32(
        /*neg_a=*/false, afrag, /*neg_b=*/false, b,
        /*c_mod=*/(short)0, cfrag, /*reuse_a=*/false, /*reuse_b=*/false);
}

__global__ __launch_bounds__(256)
void shape_context_loss_kernel(const float* __restrict__ veh,
                               const float* __restrict__ fus,
                               float* __restrict__ out) {
    __shared__ unsigned hist[16][256];   // radix-select histogram per row
    __shared__ unsigned pcount[16][32];  // final 32-partition counts per row
    __shared__ unsigned minbits[16];     // min d2 bit pattern per row
    __shared__ int      minidx[16];      // argmin fusion index per row
    __shared__ unsigned pfx[16];         // radix-select prefix (-> exact kth value)
    __shared__ int      want[16];        // remaining rank within candidate set
    __shared__ unsigned cntlt[16];       // # strictly below threshold
    __shared__ int      needEq[16];      // tie slots at threshold
    __shared__ float    ax[16], ay[16], az[16], an2[16];
    __shared__ float    q0x[16], q0y[16], q0z[16];

    const int  tid   = threadIdx.x;
    const int  lane  = tid & 31;
    const int  wave  = tid >> 5;
    const int  row0  = blockIdx.x * 16;
    const bool hi    = lane >= 16;
    const int  ncol  = lane & 15;        // column within tile / A row index
    const int  mhalf = hi ? 8 : 0;       // C/D rows: M = r + mhalf

    if (tid < 16) {
        const float* vp = veh + 3 * (row0 + tid);
        const float x = vp[0], y = vp[1], z = vp[2];
        ax[tid] = x; ay[tid] = y; az[tid] = z;
        an2[tid] = x * x + y * y + z * z;
        minbits[tid] = 0xFFFFFFFFu;
        minidx[tid]  = 0x7FFFFFFF;
        pfx[tid] = 0u; want[tid] = TOPK_ - 1; cntlt[tid] = 0u;
    }
    for (int i = tid; i < 16 * 32;  i += 256) ((unsigned*)pcount)[i] = 0u;
    for (int i = tid; i < 16 * 256; i += 256) ((unsigned*)hist)[i]   = 0u;
    __syncthreads();

    // A fragment (16x4, lanes 0-15 hold K={0,1}, lanes 16-31 hold K={2,3})
    v2f afrag;
    afrag.x = hi ? az[ncol] : ax[ncol];
    afrag.y = hi ? 1.0f     : ay[ncol];
    // C fragment: |a_m|^2 broadcast along N
    v8f cfrag;
#pragma unroll
    for (int r = 0; r < 8; ++r) cfrag[r] = an2[mhalf + r];

    // ---- 4-level MSB radix select of the 200th-smallest d2 (exact bits) ----
#pragma unroll
    for (int level = 0; level < 4; ++level) {
        const int shift = 24 - 8 * level;

        // hoist per-row pass-invariant state out of the tile loop
        unsigned pfxtop[8], minb[8];
#pragma unroll
        for (int r = 0; r < 8; ++r) {
            pfxtop[r] = (level > 0) ? (pfx[mhalf + r] >> (shift + 8)) : 0u;
            minb[r]   = (level == 1) ? minbits[mhalf + r] : 0u;
        }

        for (int t = wave; t < NTILES; t += NWAVES) {
            const v8f d   = d2_tile(fus, t * 16, afrag, cfrag, hi, ncol);
            const int col = t * 16 + ncol;
#pragma unroll
            for (int r = 0; r < 8; ++r) {
                const int      rr = mhalf + r;
                const unsigned v  = __float_as_uint(fmaxf(d[r], 0.0f));
                if (level == 0) {
                    atomicMin(&minbits[rr], v);
                    atomicAdd(&hist[rr][(v >> shift) & 255], 1u);
                } else {
                    if ((v >> (shift + 8)) == pfxtop[r])
                        atomicAdd(&hist[rr][(v >> shift) & 255], 1u);
                    if (level == 1 && v == minb[r])
                        atomicMin((unsigned*)&minidx[rr], (unsigned)col);
                }
            }
        }
        __syncthreads();
        if (tid < 16) {  // serial 256-bucket scan per row
            unsigned run = 0; const int w = want[tid]; unsigned bsel = 255u;
            for (int b = 0; b < 256; ++b) {
                const unsigned c = hist[tid][b];
                if (run + c > (unsigned)w) { bsel = (unsigned)b; break; }
                run += c;
            }
            want[tid]   = w - (int)run;
            cntlt[tid] += run;
            pfx[tid]   |= bsel << shift;
        }
        __syncthreads();
        for (int i = tid; i < 16 * 256; i += 256) ((unsigned*)hist)[i] = 0u;
        __syncthreads();
    }

    if (tid < 16) {
        needEq[tid] = TOPK_ - (int)cntlt[tid];  // >= 1 tie slots at exact kth value
        const float* fp = fus + 3 * minidx[tid];
        q0x[tid] = fp[0]; q0y[tid] = fp[1]; q0z[tid] = fp[2];
    }
    __syncthreads();

    // ---- final sweep: accept top-200 and histogram shape-context bins ----
    const float TWO_PI_F = 6.28318530717958647692f;
    const float PI_F     = 3.14159265358979323846f;
    {
        unsigned Tr[8];
#pragma unroll
        for (int r = 0; r < 8; ++r) Tr[r] = pfx[mhalf + r];

        for (int t = wave; t < NTILES; t += NWAVES) {
            const v8f d   = d2_tile(fus, t * 16, afrag, cfrag, hi, ncol);
            const int col = t * 16 + ncol;
#pragma unroll
            for (int r = 0; r < 8; ++r) {
                const int      rr = mhalf + r;
                const unsigned v  = __float_as_uint(fmaxf(d[r], 0.0f));
                bool acc = v < Tr[r];
                if (!acc && v == Tr[r]) acc = (atomicSub(&needEq[rr], 1) > 0);
                if (acc) {
                    const float* fp = fus + 3 * col;
                    const float rx = fp[0] - q0x[rr];
                    const float ry = fp[1] - q0y[rr];
                    const float rz = fp[2] - q0z[rr];
                    const float D = sqrtf(rx * rx + ry * ry + rz * rz + 1e-7f);
                    const int db = (D >= 4.0f) ? 1 : ((D >= 1.0f) ? 0 : -1);
                    if (db >= 0) {
                        const float axy = fmodf(atan2f(ry, rx) + TWO_PI_F, TWO_PI_F);
                        const float azy = fmodf(atan2f(ry, rz) + TWO_PI_F, PI_F);
                        const int xb = (int)floorf(axy / (TWO_PI_F / 4.0f));
                        const int zb = (int)floorf(azy / (PI_F / 4.0f));
                        if (xb >= 0 && zb >= 0) {
                            const int bin = db * 16 + xb * 4 + zb;
                            if (bin >= 0 && bin < 32) atomicAdd(&pcount[rr][bin], 1u);
                        }
                    }
                }
            }
        }
    }
    __syncthreads();

    // ---- counts+1 -> L2-normalize*4 -> softmax -> out ----
    if (tid < 16) {
        float c[32]; float ss = 0.0f;
        for (int p = 0; p < 32; ++p) {
            const float cc = (float)pcount[tid][p] + 1.0f;
            c[p] = cc; ss += cc * cc;
        }
        const float inv = 4.0f / sqrtf(ss);
        float mx = -1e30f;
        for (int p = 0; p < 32; ++p) { c[p] *= inv; mx = fmaxf(mx, c[p]); }
        float se = 0.0f;
        for (int p = 0; p < 32; ++p) { c[p] = expf(c[p] - mx); se += c[p]; }
        const float rcp = 1.0f / se;
        float* o = out + (size_t)(row0 + tid) * 32;
        for (int p = 0; p < 32; ++p) o[p] = c[p] * rcp;
    }
}

extern "C" void kernel_launch(void* const* d_in, const int* in_sizes, int n_in,
                              void* d_out, int out_size, void* d_ws, size_t ws_size,
                              hipStream_t stream) {
    const float* veh = (const float*)d_in[0];  // (1024, 3) f32
    const float* fus = (const float*)d_in[1];  // (20000, 3) f32
    float* out = (float*)d_out;                // (1024, 32) f32
    (void)in_sizes; (void)n_in; (void)out_size; (void)d_ws; (void)ws_size;
    shape_context_loss_kernel<<<dim3(NQ / 16), dim3(256), 0, stream>>>(veh, fus, out);
}